// Sage_42322607735218
// MI455X (gfx1250) — compile-verified
//
#include <hip/hip_runtime.h>

#define N0   409600
#define N1   40960
#define N2   4096
#define E0   409600
#define E1   40960
#define D_IN  128
#define D_HID 256
#define D_OUT 128

typedef float v2f __attribute__((ext_vector_type(2)));
typedef float v8f __attribute__((ext_vector_type(8)));

__device__ __forceinline__ v8f wmma_f32(v2f a, v2f b, v8f c) {
  // D = A(16x4 f32) * B(4x16 f32) + C(16x16 f32)
  return __builtin_amdgcn_wmma_f32_16x16x4_f32(false, a, false, b, (short)0, c, false, false);
}

// ---------------------------------------------------------------- zero scratch
__global__ void k_zero(float* __restrict__ p, int n) {
  int i = blockIdx.x * blockDim.x + threadIdx.x;
  if (i < n) p[i] = 0.0f;
}

// ------------------------------------------------- layer0 scatter (mean aggr)
// one wave per edge: 128 floats, 4 per lane (coalesced), f32 atomics into L2
__global__ void k_scatter0(const float* __restrict__ x, const int* __restrict__ src,
                           const int* __restrict__ dst, float* __restrict__ agg,
                           float* __restrict__ cnt) {
  int wid  = (blockIdx.x * blockDim.x + threadIdx.x) >> 5;
  int lane = threadIdx.x & 31;
  if (wid >= E0) return;
  int s = src[wid], d = dst[wid];
  const float* xr = x + (size_t)s * D_IN;
  float* ar = agg + (size_t)d * D_IN;
#pragma unroll
  for (int j = 0; j < D_IN / 32; ++j) {
    int f = lane + 32 * j;
    atomicAdd(ar + f, xr[f]);
  }
  if (lane == 0) atomicAdd(cnt + d, 1.0f);
}

// ------------------------------------------------- layer1 scatter (mean aggr)
__global__ void k_scatter1(const float* __restrict__ h, const int* __restrict__ src,
                           const int* __restrict__ dst, float* __restrict__ agg,
                           float* __restrict__ cnt) {
  int wid  = (blockIdx.x * blockDim.x + threadIdx.x) >> 5;
  int lane = threadIdx.x & 31;
  if (wid >= E1) return;
  int s = src[wid], d = dst[wid];
  const float* hr = h + (size_t)s * D_HID;
  float* ar = agg + (size_t)d * D_HID;
#pragma unroll
  for (int j = 0; j < D_HID / 32; ++j) {
    int f = lane + 32 * j;
    atomicAdd(ar + f, hr[f]);
  }
  if (lane == 0) atomicAdd(cnt + d, 1.0f);
}

// ---------------- layer0 dense: h = relu(mean0 @ Wl0 + bl0 + x @ Wr0), [N1,256]
// one wave computes a 16x64 strip of h (four 16x16 WMMA accumulators):
// A operands (agg row, x row) loaded once per 64 output columns.
__global__ void k_layer0(const float* __restrict__ agg, const float* __restrict__ cnt,
                         const float* __restrict__ x, const float* __restrict__ Wl,
                         const float* __restrict__ bl, const float* __restrict__ Wr,
                         float* __restrict__ h) {
  const int wid  = (blockIdx.x * blockDim.x + threadIdx.x) >> 5;
  const int lane = threadIdx.x & 31;
  const int TN = D_HID / 64;                 // 4 strips of 64 cols
  const int tm = wid / TN, tn = wid % TN;
  if (tm >= N1 / 16) return;                 // wave-uniform
  const int mrow = tm * 16 + (lane & 15);
  const int kh   = (lane >> 4) * 2;          // 0 or 2
  const int colb = tn * 64 + (lane & 15);
  const float rinv = 1.0f / fmaxf(cnt[mrow], 1.0f);
  const float* ar = agg + (size_t)mrow * D_IN;
  const float* xr = x   + (size_t)mrow * D_IN;
  v8f acc[4] = {};
#pragma unroll 2
  for (int k = 0; k < D_IN; k += 4) {
    v2f a1, a2;
    a1[0] = ar[k + kh] * rinv;  a1[1] = ar[k + kh + 1] * rinv;
    a2[0] = xr[k + kh];         a2[1] = xr[k + kh + 1];
    const float* w0 = Wl + (size_t)(k + kh) * D_HID;
    const float* w1 = Wr + (size_t)(k + kh) * D_HID;
#pragma unroll
    for (int c = 0; c < 4; ++c) {
      const int col = colb + 16 * c;
      v2f bL, bR;
      bL[0] = w0[col]; bL[1] = w0[D_HID + col];
      bR[0] = w1[col]; bR[1] = w1[D_HID + col];
      acc[c] = wmma_f32(a1, bL, acc[c]);
      acc[c] = wmma_f32(a2, bR, acc[c]);
    }
  }
  const int mbase = tm * 16 + ((lane >> 4) << 3);
#pragma unroll
  for (int c = 0; c < 4; ++c) {
    const int col = colb + 16 * c;
    const float bb = bl[col];
#pragma unroll
    for (int r = 0; r < 8; ++r)
      h[(size_t)(mbase + r) * D_HID + col] = fmaxf(acc[c][r] + bb, 0.0f);
  }
}

// ---------------- layer1 dense: z = mean1 @ Wl1 + bl1 + h[:N2] @ Wr1, [N2,128]
// small GEMM: keep 16x32 strips for wave-count/occupancy (1024 waves)
__global__ void k_layer1(const float* __restrict__ agg, const float* __restrict__ cnt,
                         const float* __restrict__ h, const float* __restrict__ Wl,
                         const float* __restrict__ bl, const float* __restrict__ Wr,
                         float* __restrict__ z) {
  const int wid  = (blockIdx.x * blockDim.x + threadIdx.x) >> 5;
  const int lane = threadIdx.x & 31;
  const int TN = D_OUT / 32;                 // 4 strips
  const int tm = wid / TN, tn = wid % TN;
  if (tm >= N2 / 16) return;
  const int mrow = tm * 16 + (lane & 15);
  const int kh   = (lane >> 4) * 2;
  const int col0 = tn * 32 + (lane & 15);
  const int col1 = col0 + 16;
  const float rinv = 1.0f / fmaxf(cnt[mrow], 1.0f);
  const float* ar = agg + (size_t)mrow * D_HID;
  const float* hr = h   + (size_t)mrow * D_HID;
  v8f acc0 = {}; v8f acc1 = {};
#pragma unroll 4
  for (int k = 0; k < D_HID; k += 4) {
    v2f a1, a2, b0l, b1l, b0r, b1r;
    a1[0] = ar[k + kh] * rinv;  a1[1] = ar[k + kh + 1] * rinv;
    a2[0] = hr[k + kh];         a2[1] = hr[k + kh + 1];
    const float* w0 = Wl + (size_t)(k + kh) * D_OUT;
    const float* w1 = Wr + (size_t)(k + kh) * D_OUT;
    b0l[0] = w0[col0]; b0l[1] = w0[D_OUT + col0];
    b1l[0] = w0[col1]; b1l[1] = w0[D_OUT + col1];
    b0r[0] = w1[col0]; b0r[1] = w1[D_OUT + col0];
    b1r[0] = w1[col1]; b1r[1] = w1[D_OUT + col1];
    acc0 = wmma_f32(a1, b0l, acc0);
    acc0 = wmma_f32(a2, b0r, acc0);
    acc1 = wmma_f32(a1, b1l, acc1);
    acc1 = wmma_f32(a2, b1r, acc1);
  }
  const int mbase = tm * 16 + ((lane >> 4) << 3);
  const float bb0 = bl[col0], bb1 = bl[col1];
#pragma unroll
  for (int r = 0; r < 8; ++r) {
    z[(size_t)(mbase + r) * D_OUT + col0] = acc0[r] + bb0;
    z[(size_t)(mbase + r) * D_OUT + col1] = acc1[r] + bb1;
  }
}

// ---------------------------------------- log_softmax rows: one wave32 per row
__global__ void k_logsoftmax(const float* __restrict__ z, float* __restrict__ out) {
  int wid  = (blockIdx.x * blockDim.x + threadIdx.x) >> 5;
  int lane = threadIdx.x & 31;
  if (wid >= N2) return;
  const float* zr = z + (size_t)wid * D_OUT;
  float v[4];
  float m = -3.402823466e38f;
#pragma unroll
  for (int j = 0; j < 4; ++j) { v[j] = zr[lane + 32 * j]; m = fmaxf(m, v[j]); }
#pragma unroll
  for (int o = 16; o > 0; o >>= 1) m = fmaxf(m, __shfl_xor(m, o, 32));
  float s = 0.0f;
#pragma unroll
  for (int j = 0; j < 4; ++j) s += __expf(v[j] - m);
#pragma unroll
  for (int o = 16; o > 0; o >>= 1) s += __shfl_xor(s, o, 32);
  float lse = m + __logf(s);
#pragma unroll
  for (int j = 0; j < 4; ++j) out[(size_t)wid * D_OUT + lane + 32 * j] = v[j] - lse;
}

// ----------------------- prod = z @ z.T : one wave per 16x64 strip, NT stores
// four accumulators: 5 b64 loads per 4 WMMA
__global__ void k_zzt(const float* __restrict__ z, float* __restrict__ prod) {
  const int wid  = (blockIdx.x * blockDim.x + threadIdx.x) >> 5;
  const int lane = threadIdx.x & 31;
  const int TN = N2 / 64;                    // 64 strips per tile-row
  const int tm = wid / TN, tn = wid % TN;
  if (tm >= N2 / 16) return;
  const int mrow = tm * 16 + (lane & 15);
  const int kh   = (lane >> 4) * 2;
  const int nb   = tn * 64 + (lane & 15);
  const float* za = z + (size_t)mrow * D_OUT;
  const float* zb[4];
#pragma unroll
  for (int c = 0; c < 4; ++c) zb[c] = z + (size_t)(nb + 16 * c) * D_OUT;
  v8f acc[4] = {};
#pragma unroll 4
  for (int k = 0; k < D_OUT; k += 4) {
    v2f a;
    a[0] = za[k + kh]; a[1] = za[k + kh + 1];
#pragma unroll
    for (int c = 0; c < 4; ++c) {
      v2f b;
      b[0] = zb[c][k + kh]; b[1] = zb[c][k + kh + 1];
      acc[c] = wmma_f32(a, b, acc[c]);
    }
  }
  const int mbase = tm * 16 + ((lane >> 4) << 3);
#pragma unroll
  for (int c = 0; c < 4; ++c) {
    const int n = nb + 16 * c;
#pragma unroll
    for (int r = 0; r < 8; ++r)
      __builtin_nontemporal_store(acc[c][r], prod + (size_t)(mbase + r) * N2 + n);
  }
}

extern "C" void kernel_launch(void* const* d_in, const int* in_sizes, int n_in,
                              void* d_out, int out_size, void* d_ws, size_t ws_size,
                              hipStream_t stream) {
  const float* x   = (const float*)d_in[0];
  const float* Wl0 = (const float*)d_in[1];
  const float* bl0 = (const float*)d_in[2];
  const float* Wr0 = (const float*)d_in[3];
  const float* Wl1 = (const float*)d_in[4];
  const float* bl1 = (const float*)d_in[5];
  const float* Wr1 = (const float*)d_in[6];
  const int* src0  = (const int*)d_in[7];
  const int* dst0  = (const int*)d_in[8];
  const int* src1  = (const int*)d_in[9];
  const int* dst1  = (const int*)d_in[10];
  float* out = (float*)d_out;

  // workspace layout (floats): agg0 | cnt0 | agg1 | cnt1 | h | z
  float* agg0 = (float*)d_ws;
  float* cnt0 = agg0 + (size_t)N1 * D_IN;
  float* agg1 = cnt0 + N1;
  float* cnt1 = agg1 + (size_t)N2 * D_HID;
  float* h    = cnt1 + N2;
  float* zbuf = h + (size_t)N1 * D_HID;

  const int zero_n = N1 * D_IN + N1 + N2 * D_HID + N2;  // accumulators + counts
  k_zero<<<(zero_n + 255) / 256, 256, 0, stream>>>(agg0, zero_n);

  // layer 0
  k_scatter0<<<(E0 * 32) / 256, 256, 0, stream>>>(x, src0, dst0, agg0, cnt0);
  {
    const int waves = (N1 / 16) * (D_HID / 64);          // 10240
    k_layer0<<<(waves * 32) / 256, 256, 0, stream>>>(agg0, cnt0, x, Wl0, bl0, Wr0, h);
  }

  // layer 1
  k_scatter1<<<(E1 * 32) / 256, 256, 0, stream>>>(h, src1, dst1, agg1, cnt1);
  {
    const int waves = (N2 / 16) * (D_OUT / 32);          // 1024
    k_layer1<<<(waves * 32) / 256, 256, 0, stream>>>(agg1, cnt1, h, Wl1, bl1, Wr1, zbuf);
  }

  // outputs
  k_logsoftmax<<<(N2 * 32) / 256, 256, 0, stream>>>(zbuf, out);
  {
    const int waves = (N2 / 16) * (N2 / 64);             // 16384
    k_zzt<<<(waves * 32) / 256, 256, 0, stream>>>(zbuf, out + (size_t)N2 * D_OUT);
  }
}